// ClusterAssignment_27041114095907
// MI455X (gfx1250) — compile-verified
//
#include <hip/hip_runtime.h>
#include <hip/hip_bf16.h>

typedef __attribute__((ext_vector_type(2))) float v2f;
typedef __attribute__((ext_vector_type(8))) float v8f;

#define N_TOTAL 65536
#define KCLUST  64
#define DFEAT   128
#define CPAD    132              // padded LDS row stride (floats) -> conflict-free ds_load_b64
#define ROWS_PER_WAVE  32        // 2 M-tiles per wave, sharing B fragments
#define ROWS_PER_BLOCK 256       // 8 waves * 32 rows
#define THREADS 256

__global__ __launch_bounds__(THREADS)
void ClusterAssignment_kernel(const float* __restrict__ z,
                              const float* __restrict__ cent,
                              float* __restrict__ q)
{
    __shared__ float ldsC[KCLUST * CPAD];   // centroids, padded rows
    __shared__ float ldsCn[KCLUST];         // ||c_j||^2

    const int tid = threadIdx.x;

    // ---- stage centroids into LDS (coalesced float4 global loads) ----
    {
        const float4* src = (const float4*)cent;
        for (int i = tid; i < (KCLUST * DFEAT) / 4; i += THREADS) {
            float4 v = src[i];
            int row = (i * 4) / DFEAT;
            int col = (i * 4) % DFEAT;
            float* d = &ldsC[row * CPAD + col];
            d[0] = v.x; d[1] = v.y; d[2] = v.z; d[3] = v.w;
        }
    }
    __syncthreads();

    // ---- per-cluster squared norms ----
    if (tid < KCLUST) {
        float s = 0.f;
        const float* row = &ldsC[tid * CPAD];
        #pragma unroll
        for (int i = 0; i < DFEAT; i += 4) {
            float a = row[i], b = row[i+1], c = row[i+2], d = row[i+3];
            s += a*a + b*b + c*c + d*d;
        }
        ldsCn[tid] = s;
    }
    __syncthreads();

    const int wave = tid >> 5;
    const int lane = tid & 31;
    const int half = lane >> 4;       // 0: lanes 0-15, 1: lanes 16-31
    const int mrow = lane & 15;
    const int rowA = blockIdx.x * ROWS_PER_BLOCK + wave * ROWS_PER_WAVE;

    const float* zrow0 = z + (size_t)(rowA + mrow) * DFEAT;        // M-tile 0
    const float* zrow1 = zrow0 + (size_t)16 * DFEAT;               // M-tile 1

    v8f acc[2][4];
    #pragma unroll
    for (int t = 0; t < 2; ++t)
        #pragma unroll
        for (int n = 0; n < 4; ++n)
            acc[t][n] = (v8f){};

    float znp0 = 0.f, znp1 = 0.f;

    // software-pipelined A fragments (prefetch 1 step ahead)
    v2f a0c = *(const v2f*)(zrow0 + 2 * half);
    v2f a1c = *(const v2f*)(zrow1 + 2 * half);

    // ---- GEMM: dot(z_i, c_j) via V_WMMA_F32_16X16X4_F32, 32 k-steps over D=128 ----
    #pragma unroll
    for (int step = 0; step < DFEAT / 4; ++step) {
        const int k0 = step * 4 + 2 * half;

        v2f a0n = a0c, a1n = a1c;
        if (step < DFEAT / 4 - 1) {               // folds at compile time
            const int kn = (step + 1) * 4 + 2 * half;
            a0n = *(const v2f*)(zrow0 + kn);
            a1n = *(const v2f*)(zrow1 + kn);
        }

        // B: 4x16 f32 tiles (= C^T), shared by both M-tiles
        v2f b0 = *(const v2f*)(&ldsC[( 0 + mrow) * CPAD + k0]);
        v2f b1 = *(const v2f*)(&ldsC[(16 + mrow) * CPAD + k0]);
        v2f b2 = *(const v2f*)(&ldsC[(32 + mrow) * CPAD + k0]);
        v2f b3 = *(const v2f*)(&ldsC[(48 + mrow) * CPAD + k0]);

        acc[0][0] = __builtin_amdgcn_wmma_f32_16x16x4_f32(false, a0c, false, b0, (short)0, acc[0][0], false, false);
        acc[0][1] = __builtin_amdgcn_wmma_f32_16x16x4_f32(false, a0c, false, b1, (short)0, acc[0][1], false, false);
        acc[0][2] = __builtin_amdgcn_wmma_f32_16x16x4_f32(false, a0c, false, b2, (short)0, acc[0][2], false, false);
        acc[0][3] = __builtin_amdgcn_wmma_f32_16x16x4_f32(false, a0c, false, b3, (short)0, acc[0][3], false, false);
        acc[1][0] = __builtin_amdgcn_wmma_f32_16x16x4_f32(false, a1c, false, b0, (short)0, acc[1][0], false, false);
        acc[1][1] = __builtin_amdgcn_wmma_f32_16x16x4_f32(false, a1c, false, b1, (short)0, acc[1][1], false, false);
        acc[1][2] = __builtin_amdgcn_wmma_f32_16x16x4_f32(false, a1c, false, b2, (short)0, acc[1][2], false, false);
        acc[1][3] = __builtin_amdgcn_wmma_f32_16x16x4_f32(false, a1c, false, b3, (short)0, acc[1][3], false, false);

        znp0 += a0c.x * a0c.x + a0c.y * a0c.y;
        znp1 += a1c.x * a1c.x + a1c.y * a1c.y;

        a0c = a0n;
        a1c = a1n;
    }

    // per-lane cluster norms for its 4 output columns
    float cn0 = ldsCn[ 0 + mrow];
    float cn1 = ldsCn[16 + mrow];
    float cn2 = ldsCn[32 + mrow];
    float cn3 = ldsCn[48 + mrow];

    // ---- epilogue per M-tile: diff -> 1/(1+diff) -> row-normalize -> store ----
    #pragma unroll
    for (int t = 0; t < 2; ++t) {
        float znp = (t == 0) ? znp0 : znp1;
        float zn = znp + __shfl_xor(znp, 16, 32);   // full ||z_row||^2, row = tilebase + mrow

        #pragma unroll
        for (int r = 0; r < 8; ++r) {
            const int m = r + 8 * half;             // C/D layout: VGPR r -> M = r + 8*half
            float znr = __shfl(zn, m, 32);

            float n0 = __builtin_amdgcn_rcpf(1.0f + (znr - 2.0f * acc[t][0][r] + cn0));
            float n1 = __builtin_amdgcn_rcpf(1.0f + (znr - 2.0f * acc[t][1][r] + cn1));
            float n2 = __builtin_amdgcn_rcpf(1.0f + (znr - 2.0f * acc[t][2][r] + cn2));
            float n3 = __builtin_amdgcn_rcpf(1.0f + (znr - 2.0f * acc[t][3][r] + cn3));

            // row sum over all 64 clusters: 4 tiles per lane + 16 lanes within half
            float part = n0 + n1 + n2 + n3;
            part += __shfl_xor(part, 1, 32);
            part += __shfl_xor(part, 2, 32);
            part += __shfl_xor(part, 4, 32);
            part += __shfl_xor(part, 8, 32);
            float inv = __builtin_amdgcn_rcpf(part);

            float* out = q + (size_t)(rowA + t * 16 + m) * KCLUST + mrow;
            out[ 0] = n0 * inv;
            out[16] = n1 * inv;
            out[32] = n2 * inv;
            out[48] = n3 * inv;
        }
    }
}

extern "C" void kernel_launch(void* const* d_in, const int* in_sizes, int n_in,
                              void* d_out, int out_size, void* d_ws, size_t ws_size,
                              hipStream_t stream) {
    const float* z    = (const float*)d_in[0];
    const float* cent = (const float*)d_in[1];
    float* q          = (float*)d_out;

    dim3 grid(N_TOTAL / ROWS_PER_BLOCK);   // 256 blocks
    dim3 block(THREADS);                   // 8 waves of 32
    ClusterAssignment_kernel<<<grid, block, 0, stream>>>(z, cent, q);
}